// GConvLSTM_44985487458637
// MI455X (gfx1250) — compile-verified
//
#include <hip/hip_runtime.h>
#include <math.h>

#define NNODES 50000
#define NEDGES 800000
#define FDIM   64
#define NF     (NNODES * FDIM)
#define NTILES (NNODES / 16)   // 3125, N is a multiple of 16

typedef __attribute__((ext_vector_type(2))) float v2f;
typedef __attribute__((ext_vector_type(8))) float v8f;

// ---------------------------------------------------------------------------
// Utility kernels
// ---------------------------------------------------------------------------
__global__ void zero_f32_kernel(float* __restrict__ p, int n) {
    int i = blockIdx.x * blockDim.x + threadIdx.x;
    if (i < n) p[i] = 0.0f;
}

__global__ void deg_kernel(const long long* __restrict__ row,
                           const float* __restrict__ w,
                           float* __restrict__ deg) {
    int e = blockIdx.x * blockDim.x + threadIdx.x;
    if (e >= NEDGES) return;
    atomicAdd(&deg[(int)row[e]], w[e]);
}

__global__ void norm_kernel(const long long* __restrict__ row,
                            const long long* __restrict__ col,
                            const float* __restrict__ w,
                            const float* __restrict__ deg,
                            float* __restrict__ nrm) {
    int e = blockIdx.x * blockDim.x + threadIdx.x;
    if (e >= NEDGES) return;
    float dr = deg[(int)row[e]];
    float dc = deg[(int)col[e]];
    float ir = dr > 0.0f ? rsqrtf(dr) : 0.0f;
    float ic = dc > 0.0f ? rsqrtf(dc) : 0.0f;
    nrm[e] = -ir * w[e] * ic;
}

// Sparse propagate: out[row[e], :] += nrm[e] * x[col[e], :]  (out pre-zeroed)
// One 32-lane wave per edge, each lane handles 2 consecutive features.
__global__ void spmm_scatter_kernel(const long long* __restrict__ row,
                                    const long long* __restrict__ col,
                                    const float* __restrict__ nrm,
                                    const float* __restrict__ x,
                                    float* __restrict__ out) {
    long long gid = (long long)blockIdx.x * blockDim.x + threadIdx.x;
    int e = (int)(gid >> 5);
    if (e >= NEDGES) return;
    int lane = (int)(gid & 31);
    int r = (int)row[e];
    int c = (int)col[e];
    float nv = nrm[e];
    const float* xp = x + (size_t)c * FDIM + lane * 2;
    float* op = out + (size_t)r * FDIM + lane * 2;
    atomicAdd(op + 0, nv * xp[0]);
    atomicAdd(op + 1, nv * xp[1]);
}

// T2 = 2*T2 - X   (Chebyshev recursion combine, in place)
__global__ void cheb_combine_kernel(float* __restrict__ t2,
                                    const float* __restrict__ x0) {
    int i = blockIdx.x * blockDim.x + threadIdx.x;
    if (i < NF) t2[i] = 2.0f * t2[i] - x0[i];
}

// ---------------------------------------------------------------------------
// Fused Chebyshev GEMM:  out = bx + bh + sum_k A_k @ Wx[k] (+ sum_k H_k @ Wh[k])
//                        (+ base).   All GEMMs are (N x 64) @ (64 x 64) f32.
// One wave computes a 16-row x 64-col output tile with V_WMMA_F32_16X16X4_F32.
// ---------------------------------------------------------------------------
__device__ __forceinline__ void wmma_accum_term(v8f acc[4],
                                                const float* __restrict__ A,
                                                const float* __restrict__ W,
                                                int rowBase, int ml, int hi) {
    const float* arow = A + (size_t)(rowBase + ml) * FDIM;
#pragma unroll
    for (int k0 = 0; k0 < FDIM; k0 += 4) {
        // 16x4 f32 A fragment: lane L holds row ml, K = k0 + 2*hi {, +1}
        v2f a;
        a.x = arow[k0 + 2 * hi + 0];
        a.y = arow[k0 + 2 * hi + 1];
#pragma unroll
        for (int nt = 0; nt < 4; ++nt) {
            // 4x16 f32 B fragment: lane L holds col ml, K = k0 + 2*hi {, +1}
            v2f b;
            b.x = W[(k0 + 2 * hi + 0) * FDIM + nt * 16 + ml];
            b.y = W[(k0 + 2 * hi + 1) * FDIM + nt * 16 + ml];
            acc[nt] = __builtin_amdgcn_wmma_f32_16x16x4_f32(
                false, a, false, b, (short)0, acc[nt], false, false);
        }
    }
}

__global__ __launch_bounds__(256)
void cheb_gemm_kernel(const float* __restrict__ A0, const float* __restrict__ A1,
                      const float* __restrict__ A2,
                      const float* __restrict__ H0, const float* __restrict__ H1,
                      const float* __restrict__ H2,
                      const float* __restrict__ Wx, const float* __restrict__ Wh,
                      const float* __restrict__ bx, const float* __restrict__ bh,
                      const float* __restrict__ base,
                      float* __restrict__ out) {
    const int lane = threadIdx.x & 31;
    const int wid  = threadIdx.x >> 5;
    const int tile = blockIdx.x * 8 + wid;
    if (tile >= NTILES) return;          // wave-uniform: EXEC all-1s for WMMA
    const int rowBase = tile * 16;
    const int ml = lane & 15;
    const int hi = lane >> 4;

    v8f acc[4];
#pragma unroll
    for (int nt = 0; nt < 4; ++nt) {
        float bv = bx[nt * 16 + ml] + (bh ? bh[nt * 16 + ml] : 0.0f);
#pragma unroll
        for (int v = 0; v < 8; ++v) acc[nt][v] = bv;
    }

    wmma_accum_term(acc, A0, Wx + 0 * FDIM * FDIM, rowBase, ml, hi);
    wmma_accum_term(acc, A1, Wx + 1 * FDIM * FDIM, rowBase, ml, hi);
    wmma_accum_term(acc, A2, Wx + 2 * FDIM * FDIM, rowBase, ml, hi);
    if (H0) {
        wmma_accum_term(acc, H0, Wh + 0 * FDIM * FDIM, rowBase, ml, hi);
        wmma_accum_term(acc, H1, Wh + 1 * FDIM * FDIM, rowBase, ml, hi);
        wmma_accum_term(acc, H2, Wh + 2 * FDIM * FDIM, rowBase, ml, hi);
    }

#pragma unroll
    for (int nt = 0; nt < 4; ++nt) {
#pragma unroll
        for (int v = 0; v < 8; ++v) {
            int r = rowBase + v + 8 * hi;     // C/D layout: VGPR v -> row v / v+8
            int c = nt * 16 + ml;
            float val = acc[nt][v];
            if (base) val += base[(size_t)r * FDIM + c];
            out[(size_t)r * FDIM + c] = val;
        }
    }
}

// ---------------------------------------------------------------------------
// Final gating: I=sig(Ipre+w_ci*C+b_i), F=sig(Fpre+w_cf*C+b_f), T=tanh(Tpre+b_c)
// out[0:NF] = H (unchanged), out[NF:2NF] = F*C + I*T
// ---------------------------------------------------------------------------
__global__ void gate_kernel(const float* __restrict__ Ipre,
                            const float* __restrict__ Fpre,
                            const float* __restrict__ Tpre,
                            const float* __restrict__ C,
                            const float* __restrict__ Hin,
                            const float* __restrict__ w_ci,
                            const float* __restrict__ w_cf,
                            const float* __restrict__ b_i,
                            const float* __restrict__ b_f,
                            const float* __restrict__ b_c,
                            float* __restrict__ out) {
    int idx = blockIdx.x * blockDim.x + threadIdx.x;
    if (idx >= NF) return;
    int f = idx & (FDIM - 1);
    float c  = C[idx];
    float iv = Ipre[idx] + w_ci[idx] * c + b_i[f];
    float fv = Fpre[idx] + w_cf[idx] * c + b_f[f];
    float tv = Tpre[idx] + b_c[f];
    float I  = 1.0f / (1.0f + expf(-iv));
    float F  = 1.0f / (1.0f + expf(-fv));
    float T  = tanhf(tv);
    out[idx]      = Hin[idx];
    out[NF + idx] = F * c + I * T;
}

// ---------------------------------------------------------------------------
extern "C" void kernel_launch(void* const* d_in, const int* in_sizes, int n_in,
                              void* d_out, int out_size, void* d_ws, size_t ws_size,
                              hipStream_t stream) {
    (void)in_sizes; (void)n_in; (void)out_size; (void)ws_size;
    const float*     X    = (const float*)d_in[0];
    const long long* ei   = (const long long*)d_in[1];
    const float*     ew   = (const float*)d_in[2];
    const float*     H    = (const float*)d_in[3];
    const float*     C    = (const float*)d_in[4];
    const float*     W_xi = (const float*)d_in[5];
    const float*     b_xi = (const float*)d_in[6];
    const float*     W_hi = (const float*)d_in[7];
    const float*     b_hi = (const float*)d_in[8];
    const float*     W_xf = (const float*)d_in[9];
    const float*     b_xf = (const float*)d_in[10];
    const float*     W_hf = (const float*)d_in[11];
    const float*     b_hf = (const float*)d_in[12];
    const float*     W_xc = (const float*)d_in[13];
    const float*     b_xc = (const float*)d_in[14];
    const float*     w_ci = (const float*)d_in[15];
    const float*     b_i  = (const float*)d_in[16];
    const float*     w_cf = (const float*)d_in[17];
    const float*     b_f  = (const float*)d_in[18];
    const float*     b_c  = (const float*)d_in[19];

    const long long* row = ei;           // edge_index[0]
    const long long* col = ei + NEDGES;  // edge_index[1]

    float* ws   = (float*)d_ws;
    float* deg  = ws;  ws += NNODES;
    float* nrm  = ws;  ws += NEDGES;
    float* PX1  = ws;  ws += NF;   // Lhat @ X
    float* TX2  = ws;  ws += NF;   // 2 Lhat PX1 - X
    float* PH1  = ws;  ws += NF;
    float* TH2  = ws;  ws += NF;
    float* Ipre = ws;  ws += NF;
    float* Fpre = ws;  ws += NF;
    float* T0   = ws;  ws += NF;
    // X/H propagations are dead after the gate GEMMs — reuse for T-branch:
    float* PT1  = PX1;
    float* TT2  = TX2;
    float* Tpre = PH1;

    const int TB = 256;
    const dim3 blk(TB);
    const dim3 gE((NEDGES + TB - 1) / TB);
    const dim3 gN((NNODES + TB - 1) / TB);
    const dim3 gNF((NF + TB - 1) / TB);
    const dim3 gSp(((long long)NEDGES * 32 + TB - 1) / TB);
    const dim3 gGemm((NTILES + 7) / 8);

    // 1) edge normalization
    zero_f32_kernel<<<gN, blk, 0, stream>>>(deg, NNODES);
    deg_kernel<<<gE, blk, 0, stream>>>(row, ew, deg);
    norm_kernel<<<gE, blk, 0, stream>>>(row, col, ew, deg, nrm);

    // 2) shared Chebyshev propagations of X and H
    zero_f32_kernel<<<gNF, blk, 0, stream>>>(PX1, NF);
    spmm_scatter_kernel<<<gSp, blk, 0, stream>>>(row, col, nrm, X, PX1);
    zero_f32_kernel<<<gNF, blk, 0, stream>>>(TX2, NF);
    spmm_scatter_kernel<<<gSp, blk, 0, stream>>>(row, col, nrm, PX1, TX2);
    cheb_combine_kernel<<<gNF, blk, 0, stream>>>(TX2, X);

    zero_f32_kernel<<<gNF, blk, 0, stream>>>(PH1, NF);
    spmm_scatter_kernel<<<gSp, blk, 0, stream>>>(row, col, nrm, H, PH1);
    zero_f32_kernel<<<gNF, blk, 0, stream>>>(TH2, NF);
    spmm_scatter_kernel<<<gSp, blk, 0, stream>>>(row, col, nrm, PH1, TH2);
    cheb_combine_kernel<<<gNF, blk, 0, stream>>>(TH2, H);

    // 3) fused 6-term WMMA GEMMs for the input / forget gate pre-activations
    cheb_gemm_kernel<<<gGemm, blk, 0, stream>>>(X, PX1, TX2, H, PH1, TH2,
                                                W_xi, W_hi, b_xi, b_hi,
                                                nullptr, Ipre);
    cheb_gemm_kernel<<<gGemm, blk, 0, stream>>>(X, PX1, TX2, H, PH1, TH2,
                                                W_xf, W_hf, b_xf, b_hf,
                                                nullptr, Fpre);

    // 4) cell candidate: T0 = cheb(X, W_xc) + b_xc
    cheb_gemm_kernel<<<gGemm, blk, 0, stream>>>(X, PX1, TX2,
                                                nullptr, nullptr, nullptr,
                                                W_xc, nullptr, b_xc, nullptr,
                                                nullptr, T0);

    // 5) T-branch propagations (reuse PX1/TX2 buffers)
    zero_f32_kernel<<<gNF, blk, 0, stream>>>(PT1, NF);
    spmm_scatter_kernel<<<gSp, blk, 0, stream>>>(row, col, nrm, T0, PT1);
    zero_f32_kernel<<<gNF, blk, 0, stream>>>(TT2, NF);
    spmm_scatter_kernel<<<gSp, blk, 0, stream>>>(row, col, nrm, PT1, TT2);
    cheb_combine_kernel<<<gNF, blk, 0, stream>>>(TT2, T0);

    // 6) Tpre = T0 + cheb(T0, W_hf) + b_hf
    cheb_gemm_kernel<<<gGemm, blk, 0, stream>>>(T0, PT1, TT2,
                                                nullptr, nullptr, nullptr,
                                                W_hf, nullptr, b_hf, nullptr,
                                                T0, Tpre);

    // 7) gates + outputs (H copy + C_new)
    gate_kernel<<<gNF, blk, 0, stream>>>(Ipre, Fpre, Tpre, C, H,
                                         w_ci, w_cf, b_i, b_f, b_c,
                                         (float*)d_out);
}